// Block_attention_guide_56968446214465
// MI455X (gfx1250) — compile-verified
//
#include <hip/hip_runtime.h>

// ============================================================================
// MI455X fused graph-transformer block (B=4096, J=17, C=512, K=4, H=256)
//
// Bandwidth-bound (~300 MB mandatory HBM traffic vs ~3.4e11 FLOPs @ 23.3TB/s
// => ~13us floor; compute is ~30x below that even at fp32-WMMA rates), so:
// fp32 everywhere (bit-faithful to reference), V_WMMA_F32_16X16X4_F32 for all
// five GEMMs, one workgroup (8 wave32) per batch fully fused in LDS (~243 KB
// of the 320 KB WGP), weights stay L2-resident, zero global intermediates.
//
// Round-3 change: the 17th-row M-tile needs no masking at all — WMMA output
// row m depends only on the A element in lane (m&15), and rows 17..31 of the
// second tile are never stored, so lanes ln!=0 may read don't-care LDS data
// (OOB LDS loads return 0 per ISA). Inner loop is now pure
// ds_load_b64 / global_load_b64 / v_wmma with linear addressing.
// ============================================================================

#define B_   4096
#define J_   17
#define C_   512
#define K_   4
#define H_   256
#define EPSF 1e-5f

typedef __attribute__((ext_vector_type(2))) float v2f;
typedef __attribute__((ext_vector_type(8))) float v8f;

__device__ __forceinline__ float gelu_exact(float x) {
  return 0.5f * x * (1.0f + erff(x * 0.70710678118654752440f));
}

// Epilogue modes
#define EP_ADD      0   // Out[m,n] += acc          (GCN passes; bias prefilled)
#define EP_GELU     1   // Out[m,n]  = gelu(acc+bias[n])
#define EP_GELU_RES 2   // Out[m,n]  = gelu(acc+bias[n]) + Res[m,n]

// GEMM: Out(17 x N) = A(17 x Kdim, LDS) * B(Kdim x N, global)
//  A addressing: A[kb*a_kb + m*a_rs + kc], kb = K-block of 512
//  B addressing: b_conv==1 -> conv1d weight layout W[(kb*512+n)*512 + kc]
//                b_conv==0 -> row-major Bg[kk*ldb + n]
__device__ __forceinline__ void gemm17(
    const float* __restrict__ A, int a_rs, int a_kb,
    const float* __restrict__ Bg, int b_conv, int ldb,
    int N, int Kdim,
    float* __restrict__ Out, int ldo,
    int ep, const float* __restrict__ bias,
    const float* __restrict__ Res, int ldr,
    int tid)
{
  const int wave = tid >> 5;
  const int lane = tid & 31;
  const int hi   = lane >> 4;   // 0|1 : K-pair select per WMMA A/B layout
  const int ln   = lane & 15;
  const int ntiles = N >> 4;

  for (int nt = wave; nt < ntiles; nt += 8) {
    const int n = (nt << 4) + ln;
    v8f acc0 = {0.f,0.f,0.f,0.f,0.f,0.f,0.f,0.f};
    v8f acc1 = {0.f,0.f,0.f,0.f,0.f,0.f,0.f,0.f};

    const int nkb = (Kdim + 511) >> 9;
    for (int kb = 0; kb < nkb; ++kb) {
      const int kcend = (Kdim - (kb << 9) >= 512) ? 512 : (Kdim - (kb << 9));
      // linear base pointers for this K-block (strength-reduced inner loop)
      const float* a0p = A + kb * a_kb + ln * a_rs + (hi << 1);
      // tile1: lane ln reads row 16+ln. Only row 16 (ln==0) is ever stored;
      // other lanes' data feed discarded output rows -> no masking needed.
      const float* a1p = A + kb * a_kb + (16 + ln) * a_rs + (hi << 1);
      const float* bkp;
      if (b_conv) bkp = Bg + ((size_t)((kb << 9) + n) << 9) + (hi << 1);
      else        bkp = Bg + (size_t)((kb << 9) + (hi << 1)) * ldb + n;

#pragma unroll 8
      for (int kc = 0; kc < kcend; kc += 4) {
        // ---- B fragment: VGPR0 = K {0|2}, VGPR1 = K {1|3}, N = lane&15 ----
        v2f bf;
        if (b_conv) {
          bf = *(const v2f*)(bkp + kc);                 // adjacent c' -> b64
        } else {
          bf.x = bkp[(size_t)kc * ldb];
          bf.y = bkp[(size_t)kc * ldb + ldb];
        }
        // ---- A fragments (LDS) ----
        v2f a0 = *(const v2f*)(a0p + kc);
        v2f a1 = *(const v2f*)(a1p + kc);
        acc0 = __builtin_amdgcn_wmma_f32_16x16x4_f32(false, a0, false, bf,
                                                     (short)0, acc0, false, false);
        acc1 = __builtin_amdgcn_wmma_f32_16x16x4_f32(false, a1, false, bf,
                                                     (short)0, acc1, false, false);
      }
    }

    // ---- epilogue: C/D layout  M = r + 8*hi (tile0) / +16 (tile1), N = ln ----
#pragma unroll
    for (int r = 0; r < 8; ++r) {
      const int m = r + (hi << 3);
      const float v = acc0[r];
      if (ep == EP_ADD)       Out[m * ldo + n] += v;
      else if (ep == EP_GELU) Out[m * ldo + n]  = gelu_exact(v + bias[n]);
      else                    Out[m * ldo + n]  = gelu_exact(v + bias[n]) + Res[m * ldr + n];
      const int mm = 16 + m;
      if (mm < J_) {
        const float v1 = acc1[r];
        if (ep == EP_ADD)       Out[mm * ldo + n] += v1;
        else if (ep == EP_GELU) Out[mm * ldo + n]  = gelu_exact(v1 + bias[n]);
        else                    Out[mm * ldo + n]  = gelu_exact(v1 + bias[n]) + Res[mm * ldr + n];
      }
    }
  }
}

// y[k,w,c] = (k==0 ? s : 1) * sum_v adj[k,v,w] * src[v,c]   (float4 over c)
__device__ __forceinline__ void joint_mix(const float* __restrict__ src,
                                          const float* __restrict__ sadj,
                                          float* __restrict__ y,
                                          float scale, int tid)
{
  for (int i = tid; i < K_ * J_ * (C_ / 4); i += 256) {
    const int c4 = i & (C_ / 4 - 1);     // 0..127
    const int r  = i >> 7;               // 0..67
    const int k  = r / J_;
    const int w  = r - k * J_;
    float ax = 0.f, ay = 0.f, az = 0.f, aw = 0.f;
#pragma unroll
    for (int v = 0; v < J_; ++v) {
      const float a  = sadj[k * J_ * J_ + v * J_ + w];
      const float4 s = ((const float4*)(src + v * C_))[c4];
      ax += a * s.x; ay += a * s.y; az += a * s.z; aw += a * s.w;
    }
    if (k == 0) { ax *= scale; ay *= scale; az *= scale; aw *= scale; }
    float4 o; o.x = ax; o.y = ay; o.z = az; o.w = aw;
    ((float4*)(y + (k * J_ + w) * C_))[c4] = o;
  }
}

// dst[m,n] = sum_k cb[k*512+n] * S[k][m] * (k==0 ? s : 1)   (conv-bias folded
// through the linear joint mix: rank-1 in (rowsum(adj_k), bias_k))
__device__ __forceinline__ void bias_prefill(float* __restrict__ dst,
                                             const float* __restrict__ S,
                                             const float* __restrict__ cb,
                                             float scale, int tid)
{
  for (int i = tid; i < J_ * (C_ / 4); i += 256) {
    const int n4 = i & (C_ / 4 - 1);
    const int m  = i >> 7;
    const float s0 = S[0 * J_ + m] * scale;
    const float s1 = S[1 * J_ + m];
    const float s2 = S[2 * J_ + m];
    const float s3 = S[3 * J_ + m];
    const float4 c0 = ((const float4*)cb)[n4];
    const float4 c1 = ((const float4*)(cb + C_))[n4];
    const float4 c2 = ((const float4*)(cb + 2 * C_))[n4];
    const float4 c3 = ((const float4*)(cb + 3 * C_))[n4];
    float4 o;
    o.x = s0 * c0.x + s1 * c1.x + s2 * c2.x + s3 * c3.x;
    o.y = s0 * c0.y + s1 * c1.y + s2 * c2.y + s3 * c3.y;
    o.z = s0 * c0.z + s1 * c1.z + s2 * c2.z + s3 * c3.z;
    o.w = s0 * c0.w + s1 * c1.w + s2 * c2.w + s3 * c3.w;
    ((float4*)dst)[i] = o;
  }
}

__global__ __launch_bounds__(256)
void fused_block_kernel(
    const float* __restrict__ x,    const float* __restrict__ adj,
    const float* __restrict__ spatial_adj,
    const float* __restrict__ cw1,  const float* __restrict__ cb1, const float* __restrict__ s1p,
    const float* __restrict__ cw2,  const float* __restrict__ cb2, const float* __restrict__ s2p,
    const float* __restrict__ n1g,  const float* __restrict__ n1b,
    const float* __restrict__ n2g,  const float* __restrict__ n2b,
    const float* __restrict__ w1,   const float* __restrict__ b1,
    const float* __restrict__ w2,   const float* __restrict__ b2,
    const float* __restrict__ w3,   const float* __restrict__ b3,
    float* __restrict__ out)
{
  __shared__ float s_x  [J_ * C_];        // x[b] -> x2 (residual anchor)
  __shared__ float s_a  [J_ * C_];        // xg / g2 / h3
  __shared__ float s_b  [J_ * C_];        // g1 / xn
  __shared__ float s_y  [K_ * J_ * C_];   // mixed activations / uMLP h1,h2
  __shared__ float s_adj[K_ * J_ * J_];
  __shared__ float s_S  [K_ * J_];        // adjacency row sums (bias fold)

  const int tid = threadIdx.x;
  const int b   = blockIdx.x;
  const float s1 = s1p[0], s2 = s2p[0];

  // ---- 1. stage x[b] and the 4 adjacency kernels into LDS ----
  {
    const float4* xv = (const float4*)(x + (size_t)b * (J_ * C_));
    float4* sxv = (float4*)s_x;
    for (int i = tid; i < J_ * C_ / 4; i += 256) sxv[i] = xv[i];
    for (int i = tid; i < K_ * J_ * J_; i += 256)
      s_adj[i] = (i < J_ * J_) ? adj[(size_t)b * J_ * J_ + i]
                               : spatial_adj[i - J_ * J_];
  }
  __syncthreads();

  // ---- 2. LayerNorm over joints axis (4 columns/thread); adj row-sums ----
  for (int c4 = tid; c4 < C_ / 4; c4 += 256) {
    float4 s = {0.f,0.f,0.f,0.f}, q = {0.f,0.f,0.f,0.f};
    for (int j = 0; j < J_; ++j) {
      const float4 v = ((const float4*)(s_x + j * C_))[c4];
      s.x += v.x; s.y += v.y; s.z += v.z; s.w += v.w;
      q.x += v.x*v.x; q.y += v.y*v.y; q.z += v.z*v.z; q.w += v.w*v.w;
    }
    const float inv = 1.f / J_;
    float4 m, ri;
    m.x = s.x*inv; m.y = s.y*inv; m.z = s.z*inv; m.w = s.w*inv;
    ri.x = rsqrtf(q.x*inv - m.x*m.x + EPSF);
    ri.y = rsqrtf(q.y*inv - m.y*m.y + EPSF);
    ri.z = rsqrtf(q.z*inv - m.z*m.z + EPSF);
    ri.w = rsqrtf(q.w*inv - m.w*m.w + EPSF);
    for (int j = 0; j < J_; ++j) {
      const float g = n1g[j], bb = n1b[j];
      const float4 v = ((const float4*)(s_x + j * C_))[c4];
      float4 o;
      o.x = (v.x - m.x) * ri.x * g + bb;
      o.y = (v.y - m.y) * ri.y * g + bb;
      o.z = (v.z - m.z) * ri.z * g + bb;
      o.w = (v.w - m.w) * ri.w * g + bb;
      ((float4*)(s_a + j * C_))[c4] = o;
    }
  }
  if (tid < K_ * J_) {
    const int k = tid / J_, w = tid - k * J_;
    float s = 0.f;
    for (int v = 0; v < J_; ++v) s += s_adj[k * J_ * J_ + v * J_ + w];
    s_S[tid] = s;
  }
  __syncthreads();

  // ---- 3. GCN pass 1: mix joints, prefill folded conv bias, WMMA GEMM ----
  joint_mix(s_a, s_adj, s_y, s1, tid);
  bias_prefill(s_b, s_S, cb1, s1, tid);
  __syncthreads();
  gemm17(s_y, C_, J_ * C_, cw1, /*b_conv=*/1, 0, C_, K_ * C_,
         s_b, C_, EP_ADD, nullptr, nullptr, 0, tid);
  __syncthreads();

  // ---- 4. GCN pass 2 ----
  joint_mix(s_b, s_adj, s_y, s2, tid);
  bias_prefill(s_a, s_S, cb2, s2, tid);
  __syncthreads();
  gemm17(s_y, C_, J_ * C_, cw2, /*b_conv=*/1, 0, C_, K_ * C_,
         s_a, C_, EP_ADD, nullptr, nullptr, 0, tid);
  __syncthreads();

  // ---- 5. residual: x2 = x + g2 ----
  {
    float4* sxv = (float4*)s_x;
    const float4* sav = (const float4*)s_a;
    for (int i = tid; i < J_ * C_ / 4; i += 256) {
      float4 a = sxv[i]; const float4 g = sav[i];
      a.x += g.x; a.y += g.y; a.z += g.z; a.w += g.w;
      sxv[i] = a;
    }
  }
  __syncthreads();

  // ---- 6. LayerNorm over channels (wave32 shuffle reduction per row) ----
  {
    const int wave = tid >> 5, lane = tid & 31;
    for (int row = wave; row < J_; row += 8) {
      float s = 0.f, q = 0.f;
      for (int c4 = lane; c4 < C_ / 4; c4 += 32) {
        const float4 v = ((const float4*)(s_x + row * C_))[c4];
        s += v.x + v.y + v.z + v.w;
        q += v.x*v.x + v.y*v.y + v.z*v.z + v.w*v.w;
      }
      for (int off = 16; off > 0; off >>= 1) {
        s += __shfl_xor(s, off, 32);
        q += __shfl_xor(q, off, 32);
      }
      const float m  = s * (1.f / C_);
      const float ri = rsqrtf(q * (1.f / C_) - m * m + EPSF);
      for (int c4 = lane; c4 < C_ / 4; c4 += 32) {
        const float4 v = ((const float4*)(s_x + row * C_))[c4];
        const float4 g = ((const float4*)n2g)[c4];
        const float4 bb = ((const float4*)n2b)[c4];
        float4 o;
        o.x = (v.x - m) * ri * g.x + bb.x;
        o.y = (v.y - m) * ri * g.y + bb.y;
        o.z = (v.z - m) * ri * g.z + bb.z;
        o.w = (v.w - m) * ri * g.w + bb.w;
        ((float4*)(s_b + row * C_))[c4] = o;
      }
    }
  }
  __syncthreads();

  // ---- 7. uMLP: 512->256 GELU, 256->256 GELU + res, 256->512 GELU ----
  float* h1 = s_y;              // 17 x 256
  float* h2 = s_y + J_ * H_;    // 17 x 256
  gemm17(s_b, C_, 0, w1, 0, H_, H_, C_, h1, H_, EP_GELU,     b1, nullptr, 0,  tid);
  __syncthreads();
  gemm17(h1,  H_, 0, w2, 0, H_, H_, H_, h2, H_, EP_GELU_RES, b2, h1,      H_, tid);
  __syncthreads();
  gemm17(h2,  H_, 0, w3, 0, C_, C_, H_, s_a, C_, EP_GELU,    b3, nullptr, 0,  tid);
  __syncthreads();

  // ---- 8. out = x2 + h ----
  {
    float4* op = (float4*)(out + (size_t)b * (J_ * C_));
    const float4* sxv = (const float4*)s_x;
    const float4* sav = (const float4*)s_a;
    for (int i = tid; i < J_ * C_ / 4; i += 256) {
      const float4 a = sxv[i]; const float4 h = sav[i];
      float4 o; o.x = a.x + h.x; o.y = a.y + h.y; o.z = a.z + h.z; o.w = a.w + h.w;
      op[i] = o;
    }
  }
}

extern "C" void kernel_launch(void* const* d_in, const int* in_sizes, int n_in,
                              void* d_out, int out_size, void* d_ws, size_t ws_size,
                              hipStream_t stream) {
  (void)in_sizes; (void)n_in; (void)out_size; (void)d_ws; (void)ws_size;
  fused_block_kernel<<<B_, 256, 0, stream>>>(
      (const float*)d_in[0],  (const float*)d_in[1],  (const float*)d_in[2],
      (const float*)d_in[3],  (const float*)d_in[4],  (const float*)d_in[5],
      (const float*)d_in[6],  (const float*)d_in[7],  (const float*)d_in[8],
      (const float*)d_in[9],  (const float*)d_in[10], (const float*)d_in[11],
      (const float*)d_in[12], (const float*)d_in[13], (const float*)d_in[14],
      (const float*)d_in[15], (const float*)d_in[16], (const float*)d_in[17],
      (const float*)d_in[18],
      (float*)d_out);
}